// HyperbolicGraphNN_30124900614688
// MI455X (gfx1250) — compile-verified
//
#include <hip/hip_runtime.h>
#include <hip/hip_bf16.h>

typedef __attribute__((ext_vector_type(2))) float v2f;
typedef __attribute__((ext_vector_type(8))) float v8f;

#define DIM 128
#define LDS_STRIDE 132   // pad 128-float rows to 132 dwords: 16B-aligned, kills bank conflicts

// ---------------------------------------------------------------------------
// Kernel 0: zero msg accumulator (d_out) and degree counters (ws)
// ---------------------------------------------------------------------------
__global__ void hgnn_zero(float* __restrict__ msg, size_t msg_n,
                          float* __restrict__ deg, int deg_n) {
    size_t i = (size_t)blockIdx.x * blockDim.x + threadIdx.x;
    size_t stride = (size_t)gridDim.x * blockDim.x;
    for (size_t k = i; k < msg_n; k += stride) msg[k] = 0.0f;
    for (size_t k = i; k < (size_t)deg_n; k += stride) deg[k] = 0.0f;
}

// ---------------------------------------------------------------------------
// Kernel 1: fused logmap0 + (tangent @ W^T + b) via V_WMMA_F32_16X16X4_F32
// Block = 256 threads (8 waves), tile = 128 rows of x.
// LDS: tangent tile (128x132 f32) + full W (128x132 f32) = 132 KB of 320 KB.
// ---------------------------------------------------------------------------
__global__ void __launch_bounds__(256)
hgnn_logmap_gemm(const float* __restrict__ x, const float* __restrict__ W,
                 const float* __restrict__ b, float* __restrict__ h, int N) {
    __shared__ float tang[DIM * LDS_STRIDE];
    __shared__ float wS[DIM * LDS_STRIDE];

    const int tid  = threadIdx.x;
    const int lane = tid & 31;
    const int wave = tid >> 5;
    const int rowBase = blockIdx.x * DIM;
    const int nLane = lane & 15;             // N index within tile

    // --- preload all 8 per-lane bias values (one wait, before any WMMA work)
    float bias[8];
    #pragma unroll
    for (int nt = 0; nt < 8; ++nt) bias[nt] = b[nt * 16 + nLane];

    // --- stage W into LDS (row-major with padded stride), float4 granularity
    for (int idx = tid; idx < DIM * (DIM / 4); idx += 256) {
        int r  = idx >> 5;        // W row (output feature n)
        int c4 = idx & 31;        // float4 chunk within row
        float4 w4 = ((const float4*)W)[r * (DIM / 4) + c4];
        *(float4*)&wS[r * LDS_STRIDE + c4 * 4] = w4;
    }

    // --- logmap0: each wave computes 16 rows; 32 lanes x float4 = 128 elems/row
    for (int rr = 0; rr < 16; ++rr) {
        int r = wave * 16 + rr;
        int g = rowBase + r;
        float4 v = {0.f, 0.f, 0.f, 0.f};
        if (g < N) v = ((const float4*)x)[(size_t)g * (DIM / 4) + lane];
        float ss = v.x * v.x + v.y * v.y + v.z * v.z + v.w * v.w;
        #pragma unroll
        for (int off = 16; off > 0; off >>= 1) ss += __shfl_xor(ss, off, 32);
        float nrm   = sqrtf(ss);
        float cl    = fminf(nrm, 1.0f - 1e-7f);
        float at    = 0.5f * logf((1.0f + cl) / (1.0f - cl));   // arctanh(cl)
        float scale = at / fmaxf(nrm, 1e-15f);
        float4 t = {v.x * scale, v.y * scale, v.z * scale, v.w * scale};
        *(float4*)&tang[r * LDS_STRIDE + lane * 4] = t;
    }
    __syncthreads();

    // --- GEMM: wave w owns rows [w*16, w*16+16); loop over 8 column tiles.
    const int m0    = wave * 16;
    const int khalf = (lane < 16) ? 0 : 2;   // K sub-offset per ISA A/B layout
    const int mHalf = (lane < 16) ? 0 : 8;   // C/D row offset per ISA layout
    const int aRow  = (m0 + nLane) * LDS_STRIDE + khalf;   // A: lane indexes M

    for (int nt = 0; nt < 8; ++nt) {
        const int n0 = nt * 16;
        v8f acc;
        #pragma unroll
        for (int v = 0; v < 8; ++v) acc[v] = bias[nt];   // C starts as +b

        const int bRow = (n0 + nLane) * LDS_STRIDE + khalf;  // B: W[n][k] (W^T)
        #pragma unroll
        for (int kk = 0; kk < DIM; kk += 4) {
            v2f a = *(const v2f*)&tang[aRow + kk];
            v2f bb = *(const v2f*)&wS[bRow + kk];
            acc = __builtin_amdgcn_wmma_f32_16x16x4_f32(
                false, a, false, bb, (short)0, acc, false, false);
        }

        // scatter D: VGPR v -> row M = v + mHalf, col N = n0 + nLane
        #pragma unroll
        for (int v = 0; v < 8; ++v) {
            int g = rowBase + m0 + v + mHalf;
            if (g < N) h[(size_t)g * DIM + n0 + nLane] = acc[v];
        }
    }
}

// ---------------------------------------------------------------------------
// Kernel 2: edge scatter — one wave per edge, float atomics into msg + deg
// ---------------------------------------------------------------------------
__global__ void hgnn_edge_scatter(const float* __restrict__ h,
                                  const int* __restrict__ src,
                                  const int* __restrict__ dst,
                                  float* __restrict__ msg,
                                  float* __restrict__ deg, int E) {
    int gwave = (int)((blockIdx.x * (size_t)blockDim.x + threadIdx.x) >> 5);
    int lane  = threadIdx.x & 31;
    if (gwave >= E) return;
    int s = src[gwave];
    int d = dst[gwave];
    float4 v = ((const float4*)h)[(size_t)s * (DIM / 4) + lane];
    float* base = msg + (size_t)d * DIM + lane * 4;
    atomicAdd(base + 0, v.x);
    atomicAdd(base + 1, v.y);
    atomicAdd(base + 2, v.z);
    atomicAdd(base + 3, v.w);
    if (lane == 0) atomicAdd(deg + d, 1.0f);
}

// ---------------------------------------------------------------------------
// Kernel 3: finalize — mean + expmap0, in place on d_out. One wave per node.
// ---------------------------------------------------------------------------
__global__ void hgnn_finalize(float* __restrict__ msg,
                              const float* __restrict__ deg, int N) {
    int node = (int)((blockIdx.x * (size_t)blockDim.x + threadIdx.x) >> 5);
    int lane = threadIdx.x & 31;
    if (node >= N) return;
    float dg = fmaxf(deg[node], 1.0f);
    float inv = 1.0f / dg;
    float4 v = ((const float4*)msg)[(size_t)node * (DIM / 4) + lane];
    v.x *= inv; v.y *= inv; v.z *= inv; v.w *= inv;
    float ss = v.x * v.x + v.y * v.y + v.z * v.z + v.w * v.w;
    #pragma unroll
    for (int off = 16; off > 0; off >>= 1) ss += __shfl_xor(ss, off, 32);
    float nrm   = sqrtf(ss);
    float scale = tanhf(nrm) / fmaxf(nrm, 1e-15f);
    float4 o = {v.x * scale, v.y * scale, v.z * scale, v.w * scale};
    ((float4*)msg)[(size_t)node * (DIM / 4) + lane] = o;
}

// ---------------------------------------------------------------------------
extern "C" void kernel_launch(void* const* d_in, const int* in_sizes, int n_in,
                              void* d_out, int out_size, void* d_ws, size_t ws_size,
                              hipStream_t stream) {
    const float* x  = (const float*)d_in[0];
    const float* W  = (const float*)d_in[1];
    const float* b  = (const float*)d_in[2];
    const int*  src = (const int*)d_in[3];
    const int*  dst = (const int*)d_in[4];

    const int N = in_sizes[0] / DIM;
    const int E = in_sizes[3];

    float* msg = (float*)d_out;                 // accumulate msg_sum in d_out
    float* h   = (float*)d_ws;                  // N*128 floats
    float* deg = h + (size_t)N * DIM;           // N floats

    // 1) zero accumulators
    hgnn_zero<<<2048, 256, 0, stream>>>(msg, (size_t)N * DIM, deg, N);

    // 2) logmap0 + linear layer (fp32 WMMA)
    hgnn_logmap_gemm<<<(N + DIM - 1) / DIM, 256, 0, stream>>>(x, W, b, h, N);

    // 3) edge aggregation (one wave per edge)
    hgnn_edge_scatter<<<(E + 7) / 8, 256, 0, stream>>>(h, src, dst, msg, deg, E);

    // 4) mean + expmap0
    hgnn_finalize<<<(N + 7) / 8, 256, 0, stream>>>(msg, deg, N);
}